// InteractionBlock_71433896067582
// MI455X (gfx1250) — compile-verified
//
#include <hip/hip_runtime.h>
#include <math.h>

#define NNODE 25000
#define NEDGE 400000

typedef __attribute__((ext_vector_type(2))) float v2f;
typedef __attribute__((ext_vector_type(8))) float v8f;

// D = A(16x4,f32) * B(4x16,f32) + C(16x16,f32)   -- native CDNA5 fp32 WMMA
__device__ __forceinline__ v8f wmma4(v2f a, v2f b, v8f c) {
  return __builtin_amdgcn_wmma_f32_16x16x4_f32(false, a, false, b, (short)0, c,
                                               false, false);
}

// B fragment: lanes 0-15 hold rows k0,k0+1 ; lanes 16-31 rows k0+2,k0+3 (col n0+m)
__device__ __forceinline__ v2f loadB(const float* __restrict__ W, int ldb,
                                     int k0, int n0, int m, int kh) {
  int r = k0 + 2 * kh;
  v2f b;
  b.x = W[r * ldb + n0 + m];
  b.y = W[(r + 1) * ldb + n0 + m];
  return b;
}

// intra-wave LDS RAW fence: DS ops are in-order per wave; stop compiler
// reordering and drain DScnt so loaded values are in VGPRs.
__device__ __forceinline__ void wave_lds_fence() {
  __builtin_amdgcn_wave_barrier();
  asm volatile("s_wait_dscnt 0" ::: "memory");
  __builtin_amdgcn_wave_barrier();
}

// ---------------------------------------------------------------------------
// Kernel 1: per-node linears.  out = c_s * s  (s0/s1), ws <- h0,h1
// ---------------------------------------------------------------------------
__global__ __launch_bounds__(256) void node_pre_kernel(
    const float* __restrict__ node_input, const float* __restrict__ node_attr,
    const float* __restrict__ Wsc0, const float* __restrict__ Wsc1,
    const float* __restrict__ Wl10, const float* __restrict__ Wl11,
    float* __restrict__ out, float* __restrict__ h0w, float* __restrict__ h1w,
    int numTiles) {
  int lane = threadIdx.x & 31;
  int wv = threadIdx.x >> 5;
  int tile = blockIdx.x * 8 + wv;
  if (tile >= numTiles) return;
  int m = lane & 15, kh = lane >> 4;
  int row0 = tile * 16;
  int rowA = row0 + m;
  if (rowA >= NNODE) rowA = NNODE - 1;
  const float c_s = 0.3826834323650898f;   // sin(pi/8)
  const float inv_m = 0.125f;              // 1/sqrt(64)

  // ---------------- x0 path ----------------
  v2f afr[16];
#pragma unroll
  for (int kk = 0; kk < 16; ++kk) {
    int k = kk * 4 + 2 * kh;
    afr[kk].x = node_input[(size_t)rowA * 256 + k];
    afr[kk].y = node_input[(size_t)rowA * 256 + k + 1];
  }
#pragma unroll
  for (int nt = 0; nt < 4; ++nt) {
    v8f accS = {};
    v8f accH = {};
#pragma unroll
    for (int kk = 0; kk < 16; ++kk) {
      accS = wmma4(afr[kk], loadB(Wsc0, 64, kk * 4, nt * 16, m, kh), accS);
      accH = wmma4(afr[kk], loadB(Wl10, 64, kk * 4, nt * 16, m, kh), accH);
    }
    int col = nt * 16 + m;
#pragma unroll
    for (int j = 0; j < 8; ++j) {
      int orow = row0 + j + 8 * kh;
      if (orow < NNODE) {
        float zf = node_attr[orow] * inv_m;
        out[(size_t)orow * 256 + col] = c_s * zf * accS[j];
        h0w[(size_t)orow * 64 + col] = zf * accH[j];
      }
    }
  }
  // ---------------- x1 (3 channels) ----------------
#pragma unroll 1
  for (int c = 0; c < 3; ++c) {
#pragma unroll
    for (int kk = 0; kk < 16; ++kk) {
      int k = kk * 4 + 2 * kh;
      afr[kk].x = node_input[(size_t)rowA * 256 + 64 + k * 3 + c];
      afr[kk].y = node_input[(size_t)rowA * 256 + 64 + (k + 1) * 3 + c];
    }
#pragma unroll
    for (int nt = 0; nt < 4; ++nt) {
      v8f accS = {};
      v8f accH = {};
#pragma unroll
      for (int kk = 0; kk < 16; ++kk) {
        accS = wmma4(afr[kk], loadB(Wsc1, 64, kk * 4, nt * 16, m, kh), accS);
        accH = wmma4(afr[kk], loadB(Wl11, 64, kk * 4, nt * 16, m, kh), accH);
      }
      int col = nt * 16 + m;
#pragma unroll
      for (int j = 0; j < 8; ++j) {
        int orow = row0 + j + 8 * kh;
        if (orow < NNODE) {
          float zf = node_attr[orow] * inv_m;
          out[(size_t)orow * 256 + 64 + col * 3 + c] = c_s * zf * accS[j];
          h1w[(size_t)orow * 192 + col * 3 + c] = zf * accH[j];
        }
      }
    }
  }
}

// ---------------------------------------------------------------------------
// Kernel 2: fused edge MLP (WMMA) + gather + tensor-product + atomic scatter
// ---------------------------------------------------------------------------
__global__ __launch_bounds__(256) void edge_kernel(
    const float* __restrict__ emb, const float* __restrict__ fcw0,
    const float* __restrict__ fcw1, const int* __restrict__ esrc,
    const int* __restrict__ edst, const float* __restrict__ eattr,
    const float* __restrict__ h0w, const float* __restrict__ h1w,
    float* __restrict__ y0w, float* __restrict__ y1w, int numTiles) {
  __shared__ float smem[8 * 16 * 65];  // per-wave 16x64 tile, pad 65 (no conflicts)
  int lane = threadIdx.x & 31;
  int wv = threadIdx.x >> 5;
  int tile = blockIdx.x * 8 + wv;
  if (tile >= numTiles) return;
  float* L = smem + wv * (16 * 65);
  int m = lane & 15, kh = lane >> 4;
  int e0 = tile * 16;

  // --- GEMM1: hid = silu( emb(16x8) @ fc_w0(8x64) / sqrt(8) ) ---
  v2f aemb[2];
  {
    int eA = e0 + m;
    aemb[0].x = emb[(size_t)eA * 8 + 0 + 2 * kh];
    aemb[0].y = emb[(size_t)eA * 8 + 1 + 2 * kh];
    aemb[1].x = emb[(size_t)eA * 8 + 4 + 2 * kh];
    aemb[1].y = emb[(size_t)eA * 8 + 5 + 2 * kh];
  }
#pragma unroll
  for (int nt = 0; nt < 4; ++nt) {
    v8f acc = {};
    acc = wmma4(aemb[0], loadB(fcw0, 64, 0, nt * 16, m, kh), acc);
    acc = wmma4(aemb[1], loadB(fcw0, 64, 4, nt * 16, m, kh), acc);
#pragma unroll
    for (int j = 0; j < 8; ++j) {
      float v = acc[j] * 0.3535533905932738f;  // 1/sqrt(NB)
      L[(j + 8 * kh) * 65 + nt * 16 + m] = v / (1.0f + __expf(-v));  // silu
    }
  }
  wave_lds_fence();

  // re-load hid tile as A fragments (K = 64)
  v2f ah[16];
#pragma unroll
  for (int kk = 0; kk < 16; ++kk) {
    int k = kk * 4 + 2 * kh;
    ah[kk].x = L[m * 65 + k];
    ah[kk].y = L[m * 65 + k + 1];
  }
  wave_lds_fence();

  // per-edge metadata (lanes 0-15 and 16-31 share edge m of the tile)
  int eM = e0 + m;
  int src = esrc[eM];
  int dst = edst[eM];
  float sh0 = eattr[(size_t)eM * 4 + 0];
  float s1x = eattr[(size_t)eM * 4 + 1];
  float s1y = eattr[(size_t)eM * 4 + 2];
  float s1z = eattr[(size_t)eM * 4 + 3];
  const float* h0s = h0w + (size_t)src * 64;
  const float* h1s = h1w + (size_t)src * 192;
  float* y0d = y0w + (size_t)dst * 128;
  float* y1d = y1w + (size_t)dst * 384;
  int uh = kh * 32;  // lane-half splits the 64 u's
  const float inv_nb = 0.25f;  // 1/sqrt(16)

  // --- GEMM2 in 4 column groups (wA,wB,wC,wD), reuse LDS tile ---
#pragma unroll 1
  for (int g = 0; g < 4; ++g) {
#pragma unroll
    for (int nt = 0; nt < 4; ++nt) {
      v8f acc = {};
#pragma unroll
      for (int kk = 0; kk < 16; ++kk)
        acc = wmma4(ah[kk], loadB(fcw1, 256, kk * 4, g * 64 + nt * 16, m, kh), acc);
#pragma unroll
      for (int j = 0; j < 8; ++j)
        L[(j + 8 * kh) * 65 + nt * 16 + m] = acc[j] * 0.125f;  // 1/sqrt(RN)
    }
    wave_lds_fence();

    if (g == 0) {  // midA -> y0[:, u]
      for (int i = 0; i < 32; ++i) {
        int u = uh + i;
        atomicAdd(&y0d[u], L[m * 65 + u] * h0s[u] * sh0 * inv_nb);
      }
    } else if (g == 1) {  // midB -> y1[:, u, c]
      for (int i = 0; i < 32; ++i) {
        int u = uh + i;
        float base = L[m * 65 + u] * h0s[u] * inv_nb;
        atomicAdd(&y1d[u * 3 + 0], base * s1x);
        atomicAdd(&y1d[u * 3 + 1], base * s1y);
        atomicAdd(&y1d[u * 3 + 2], base * s1z);
      }
    } else if (g == 2) {  // midC -> y1[:, 64+u, c]
      for (int i = 0; i < 32; ++i) {
        int u = uh + i;
        float base = L[m * 65 + u] * sh0 * inv_nb;
        atomicAdd(&y1d[(64 + u) * 3 + 0], base * h1s[u * 3 + 0]);
        atomicAdd(&y1d[(64 + u) * 3 + 1], base * h1s[u * 3 + 1]);
        atomicAdd(&y1d[(64 + u) * 3 + 2], base * h1s[u * 3 + 2]);
      }
    } else {  // midD -> y0[:, 64+u]
      for (int i = 0; i < 32; ++i) {
        int u = uh + i;
        float dot = h1s[u * 3] * s1x + h1s[u * 3 + 1] * s1y + h1s[u * 3 + 2] * s1z;
        atomicAdd(&y0d[64 + u], L[m * 65 + u] * dot * 0.5773502691896258f * inv_nb);
      }
    }
    wave_lds_fence();
  }
}

// ---------------------------------------------------------------------------
// Kernel 3: out += c_x * (y @ W_lin2) * z / sqrt(128)
// ---------------------------------------------------------------------------
__global__ __launch_bounds__(256) void node_post_kernel(
    const float* __restrict__ y0w, const float* __restrict__ y1w,
    const float* __restrict__ Wl20, const float* __restrict__ Wl21,
    const float* __restrict__ node_attr, float* __restrict__ out, int numTiles) {
  int lane = threadIdx.x & 31;
  int wv = threadIdx.x >> 5;
  int tile = blockIdx.x * 8 + wv;
  if (tile >= numTiles) return;
  int m = lane & 15, kh = lane >> 4;
  int row0 = tile * 16;
  int rowA = row0 + m;
  if (rowA >= NNODE) rowA = NNODE - 1;
  const float c_x = 0.9238795325112867f;      // cos(pi/8)
  const float inv_2m = 0.0883883476483184f;   // 1/sqrt(128)

  {  // o0 = y0(16x128) @ W_lin2_0(128x64)
    v2f afr[32];
#pragma unroll
    for (int kk = 0; kk < 32; ++kk) {
      int k = kk * 4 + 2 * kh;
      afr[kk].x = y0w[(size_t)rowA * 128 + k];
      afr[kk].y = y0w[(size_t)rowA * 128 + k + 1];
    }
#pragma unroll
    for (int nt = 0; nt < 4; ++nt) {
      v8f acc = {};
#pragma unroll
      for (int kk = 0; kk < 32; ++kk)
        acc = wmma4(afr[kk], loadB(Wl20, 64, kk * 4, nt * 16, m, kh), acc);
      int col = nt * 16 + m;
#pragma unroll
      for (int j = 0; j < 8; ++j) {
        int orow = row0 + j + 8 * kh;
        if (orow < NNODE) {
          float zf = node_attr[orow] * inv_2m * c_x;
          out[(size_t)orow * 256 + col] += zf * acc[j];
        }
      }
    }
  }
#pragma unroll 1
  for (int c = 0; c < 3; ++c) {  // o1 per channel
    v2f afr[32];
#pragma unroll
    for (int kk = 0; kk < 32; ++kk) {
      int k = kk * 4 + 2 * kh;
      afr[kk].x = y1w[(size_t)rowA * 384 + k * 3 + c];
      afr[kk].y = y1w[(size_t)rowA * 384 + (k + 1) * 3 + c];
    }
#pragma unroll
    for (int nt = 0; nt < 4; ++nt) {
      v8f acc = {};
#pragma unroll
      for (int kk = 0; kk < 32; ++kk)
        acc = wmma4(afr[kk], loadB(Wl21, 64, kk * 4, nt * 16, m, kh), acc);
      int col = nt * 16 + m;
#pragma unroll
      for (int j = 0; j < 8; ++j) {
        int orow = row0 + j + 8 * kh;
        if (orow < NNODE) {
          float zf = node_attr[orow] * inv_2m * c_x;
          out[(size_t)orow * 256 + 64 + col * 3 + c] += zf * acc[j];
        }
      }
    }
  }
}

extern "C" void kernel_launch(void* const* d_in, const int* in_sizes, int n_in,
                              void* d_out, int out_size, void* d_ws,
                              size_t ws_size, hipStream_t stream) {
  (void)in_sizes; (void)n_in; (void)out_size; (void)ws_size;
  const float* node_input = (const float*)d_in[0];
  const float* node_attr  = (const float*)d_in[1];
  const int*   edge_src   = (const int*)d_in[2];
  const int*   edge_dst   = (const int*)d_in[3];
  const float* edge_attr  = (const float*)d_in[4];
  const float* emb        = (const float*)d_in[5];
  const float* Wsc0       = (const float*)d_in[6];
  const float* Wsc1       = (const float*)d_in[7];
  const float* Wl10       = (const float*)d_in[8];
  const float* Wl11       = (const float*)d_in[9];
  const float* fcw0       = (const float*)d_in[10];
  const float* fcw1       = (const float*)d_in[11];
  const float* Wl20       = (const float*)d_in[12];
  const float* Wl21       = (const float*)d_in[13];
  float* out = (float*)d_out;
  float* ws  = (float*)d_ws;

  float* h0w = ws;                          // N x 64
  float* h1w = ws + (size_t)NNODE * 64;     // N x 64 x 3
  float* y0w = ws + (size_t)NNODE * 256;    // N x 128
  float* y1w = ws + (size_t)NNODE * 384;    // N x 128 x 3

  hipMemsetAsync(y0w, 0, (size_t)NNODE * 512 * sizeof(float), stream);

  int nTilesN = (NNODE + 15) / 16;  // 1563
  int nTilesE = NEDGE / 16;         // 25000
  int blocksN = (nTilesN + 7) / 8;
  int blocksE = (nTilesE + 7) / 8;

  node_pre_kernel<<<blocksN, 256, 0, stream>>>(node_input, node_attr, Wsc0,
                                               Wsc1, Wl10, Wl11, out, h0w, h1w,
                                               nTilesN);
  edge_kernel<<<blocksE, 256, 0, stream>>>(emb, fcw0, fcw1, edge_src, edge_dst,
                                           edge_attr, h0w, h1w, y0w, y1w,
                                           nTilesE);
  node_post_kernel<<<blocksN, 256, 0, stream>>>(y0w, y1w, Wl20, Wl21, node_attr,
                                                out, nTilesN);
}